// ShiftNMF_13932873909176
// MI455X (gfx1250) — compile-verified
//
#include <hip/hip_runtime.h>
#include <hip/hip_bf16.h>

// ShiftNMF forward for MI455X (gfx1250, wave32).
//
// Stage 1: Hp = softplus(H)                       (64 x 1024 f32)
// Stage 2: Hf = DFT(Hp) via V_WMMA_F32_16X16X4_F32 (64 x 1024 complex, interleaved)
// Stage 3: WHt[n,f] = sum_r softplus(W)[n,r] * e^{-i 2pi f tau[n,r]/M} * Hf[r,f]
//          with a per-thread geometric phase recurrence over 16 consecutive f.

typedef __attribute__((ext_vector_type(2))) float v2f;
typedef __attribute__((ext_vector_type(8))) float v8f;

#define MM   1024   // frequency bins / DFT length
#define NN   1024   // rows of W
#define RANK 64

__device__ __forceinline__ float softplus_f(float x) {
    // exact for large x, log1p(exp(x)) otherwise
    return (x > 15.0f) ? x : log1pf(__expf(x));
}

// ---------------------------------------------------------------------------
// Stage 1: elementwise softplus of H into workspace
// ---------------------------------------------------------------------------
__global__ void __launch_bounds__(256)
shiftnmf_softplus(const float* __restrict__ H, float* __restrict__ Hp, int nElem) {
    int i = blockIdx.x * blockDim.x + threadIdx.x;
    if (i < nElem) Hp[i] = softplus_f(H[i]);
}

// ---------------------------------------------------------------------------
// Stage 2: Hf[r,f] = sum_m Hp[r,m] * exp(-i*2*pi*m*f/M)  via f32 WMMA.
// One wave per 16x16 output tile; 8 waves per block; 256 tiles total
// (4 rank-row tiles x 64 freq tiles). K loop: 1024/4 = 256 WMMA pairs (re, im).
// Twiddle angle uses exact integer (m*f) mod 1024 -> no f32 range reduction.
// ---------------------------------------------------------------------------
__global__ void __launch_bounds__(256)
shiftnmf_dft_wmma(const float* __restrict__ Hp, float2* __restrict__ Hf) {
    const int lane = threadIdx.x & 31;
    const int wave = threadIdx.x >> 5;
    const int tile = blockIdx.x * 8 + wave;   // 0..255
    const int r0   = (tile >> 6) * 16;        // rank-row tile base (0,16,32,48)
    const int f0   = (tile & 63) * 16;        // freq tile base

    const int half = lane >> 4;               // 0: K={0,1}, 1: K={2,3}
    const int l16  = lane & 15;
    const int arow = r0 + l16;                // A-matrix row for this lane
    const int fcol = f0 + l16;                // B-matrix column for this lane

    const float W0 = -6.28318530717958647692f / (float)MM;

    v8f acc_re = {};
    v8f acc_im = {};

    for (int m = 0; m < MM; m += 4) {
        const int kb = m + half * 2;

        // A tile 16x4 (2 VGPRs): lane holds Hp[arow, kb] and Hp[arow, kb+1]
        v2f a;
        a.x = Hp[arow * MM + kb + 0];
        a.y = Hp[arow * MM + kb + 1];

        // B tiles 4x16: twiddle factors, exact integer mod for the angle
        const int idx0 = ((kb + 0) * fcol) & (MM - 1);
        const int idx1 = ((kb + 1) * fcol) & (MM - 1);
        float s0, c0, s1, c1;
        __sincosf(W0 * (float)idx0, &s0, &c0);
        __sincosf(W0 * (float)idx1, &s1, &c1);

        v2f b_re, b_im;
        b_re.x = c0; b_re.y = c1;   // cos(-x) plane
        b_im.x = s0; b_im.y = s1;   // sin(-x) plane (angle already negative)

        acc_re = __builtin_amdgcn_wmma_f32_16x16x4_f32(
            false, a, false, b_re, (short)0, acc_re, false, false);
        acc_im = __builtin_amdgcn_wmma_f32_16x16x4_f32(
            false, a, false, b_im, (short)0, acc_im, false, false);
    }

    // C/D layout: VGPR j, lanes 0-15 -> row j, lanes 16-31 -> row j+8
#pragma unroll
    for (int j = 0; j < 8; ++j) {
        const int row = r0 + j + half * 8;
        Hf[row * MM + fcol] = make_float2(acc_re[j], acc_im[j]);
    }
}

// ---------------------------------------------------------------------------
// Stage 3: complex contraction with geometric phase recurrence.
// One block (64 threads) per output row n; each thread owns 16 consecutive f.
// Per (r, thread): 2 sincos + 16 complex FMAs + 16 complex rotations.
// |theta*f| < 2*pi (tau in [0,1)), so no range-reduction issues.
// ---------------------------------------------------------------------------
__global__ void __launch_bounds__(64)
shiftnmf_combine(const float* __restrict__ W,
                 const float* __restrict__ tau,
                 const float2* __restrict__ Hf,
                 float2* __restrict__ out) {
    const int n  = blockIdx.x;
    const int f0 = threadIdx.x * 16;

    float acc_re[16], acc_im[16];
#pragma unroll
    for (int j = 0; j < 16; ++j) { acc_re[j] = 0.0f; acc_im[j] = 0.0f; }

    const float W0 = -6.28318530717958647692f / (float)MM;

    for (int r = 0; r < RANK; ++r) {
        const float wp    = softplus_f(W[n * RANK + r]);
        const float theta = W0 * tau[n * RANK + r];

        float ss, cs, s0, c0;
        __sincosf(theta, &ss, &cs);               // unit step e^{i*theta}
        __sincosf(theta * (float)f0, &s0, &c0);   // base phase at f0

        float pr = wp * c0;                       // wp folded into the rotor
        float pi = wp * s0;

        const float2* __restrict__ hrow = Hf + r * MM + f0;
#pragma unroll
        for (int j = 0; j < 16; ++j) {
            const float2 h = hrow[j];
            acc_re[j] = fmaf(pr, h.x, fmaf(-pi, h.y, acc_re[j]));
            acc_im[j] = fmaf(pr, h.y, fmaf( pi, h.x, acc_im[j]));
            const float npr = fmaf(pr, cs, -pi * ss);
            const float npi = fmaf(pr, ss,  pi * cs);
            pr = npr; pi = npi;
        }
    }

    float2* __restrict__ orow = out + n * MM + f0;
#pragma unroll
    for (int j = 0; j < 16; ++j) orow[j] = make_float2(acc_re[j], acc_im[j]);
}

// ---------------------------------------------------------------------------
// Host-side launcher (graph-capture safe: only kernel launches on `stream`)
// ---------------------------------------------------------------------------
extern "C" void kernel_launch(void* const* d_in, const int* in_sizes, int n_in,
                              void* d_out, int out_size, void* d_ws, size_t ws_size,
                              hipStream_t stream) {
    const float* W   = (const float*)d_in[0];   // (1024, 64)
    const float* H   = (const float*)d_in[1];   // (64, 1024)
    const float* tau = (const float*)d_in[2];   // (1024, 64)

    // workspace layout: Hp (64*1024 f32) | Hf (64*1024 float2)
    float*  Hp = (float*)d_ws;
    float2* Hf = (float2*)((char*)d_ws + (size_t)RANK * MM * sizeof(float));

    float2* WHt = (float2*)d_out;               // (1024, 1024) complex64 interleaved

    const int hElems = RANK * MM;               // 65536

    shiftnmf_softplus<<<(hElems + 255) / 256, 256, 0, stream>>>(H, Hp, hElems);

    // 256 tiles (4 rank-row tiles x 64 freq tiles), 8 waves/block -> 32 blocks
    shiftnmf_dft_wmma<<<32, 256, 0, stream>>>(Hp, Hf);

    // one block per output row n
    shiftnmf_combine<<<NN, 64, 0, stream>>>(W, tau, Hf, WHt);
}